// ROIAlign3D_6957847020160
// MI455X (gfx1250) — compile-verified
//
#include <hip/hip_runtime.h>
#include <stdint.h>

#define ROI   16
#define CCH   64
#define DIMW  96
#define HW    (96 * 96)
#define DHW   (96 * 96 * 96)

// One workgroup per proposal. 8 waves; wave w handles channels [8w, 8w+8).
// Each wave double-buffers 4KB slabs ([4][16][16] floats) in LDS via
// global_load_async_to_lds_b128, overlapping DMA with max-pooling.
__global__ __launch_bounds__(256)
void roi_maxpool3d_async_kernel(const float* __restrict__ fm,
                                const int* __restrict__ props,
                                float* __restrict__ out)
{
    __shared__ float lds[8 * 2 * 1024];   // 8 waves x 2 buffers x 1024 floats = 64KB

    const int n    = blockIdx.x;
    const int tid  = threadIdx.x;
    const int wave = tid >> 5;
    const int lane = tid & 31;

    const int b  = props[n * 7 + 0];
    const int z1 = props[n * 7 + 1];
    const int y1 = props[n * 7 + 2];
    const int x1 = props[n * 7 + 3];

    // LDS byte address of this wave's staging area (low 32 bits of the
    // generic shared-pointer are the LDS offset on gfx1250).
    const unsigned ldsWaveBase =
        (unsigned)(uintptr_t)(&lds[0]) + (unsigned)(wave * 2 * 1024) * 4u;

    // Global base of the ROI corner for channel 0 of this proposal's batch.
    const float* gbase = fm + (size_t)b * CCH * (size_t)DHW
                            + (size_t)z1 * HW + (size_t)y1 * DIMW + (size_t)x1;

    const int c0 = wave * 8;

    // Issue 8 async 16B loads staging slab t (channel c0 + t/4, z-window t%4)
    // into LDS buffer `buf`. Flat slab index f = r*128 + lane*4 decomposes as
    // [dz][yy][xs] over a [4][16][16] tile.
    auto issue = [&](int t, int buf) {
        const int c  = c0 + (t >> 2);
        const int zo = t & 3;
        const float* sbase = gbase + (size_t)c * DHW + (size_t)(zo * 4) * HW;
        const unsigned ldsBuf = ldsWaveBase + (unsigned)buf * 4096u;
        #pragma unroll
        for (int r = 0; r < 8; ++r) {
            const int f  = r * 128 + lane * 4;
            const int dz = f >> 8;
            const int yy = (f >> 4) & 15;
            const int xs = f & 15;
            unsigned long long ga = (unsigned long long)(uintptr_t)
                (sbase + (size_t)dz * HW + (size_t)yy * DIMW + (size_t)xs);
            unsigned la = ldsBuf + (unsigned)f * 4u;
            asm volatile("global_load_async_to_lds_b128 %0, %1, off"
                         :: "v"(la), "v"(ga) : "memory");
        }
    };

    issue(0, 0);

    const int oi   = lane & 15;   // output element within slab: y_out*4 + x_out
    const int half = lane >> 4;   // which pair of z-planes this lane reduces
    const int yo   = oi >> 2;
    const int xo   = oi & 3;

    for (int t = 0; t < 32; ++t) {
        if (t + 1 < 32) {
            issue(t + 1, (t + 1) & 1);
            // 8 loads of slab t+1 outstanding; async loads retire in order,
            // so <=8 outstanding means slab t is fully resident in LDS.
            asm volatile("s_wait_asynccnt 8" ::: "memory");
        } else {
            asm volatile("s_wait_asynccnt 0" ::: "memory");
        }

        const float* buf = &lds[wave * 2048 + (t & 1) * 1024];

        // Each lane: max over 2 z-planes x 4 rows x 4 x-values (8x ds_load_b128).
        float4 m = make_float4(-__builtin_inff(), -__builtin_inff(),
                               -__builtin_inff(), -__builtin_inff());
        #pragma unroll
        for (int dz = 0; dz < 2; ++dz) {
            #pragma unroll
            for (int dy = 0; dy < 4; ++dy) {
                const float4 v = *(const float4*)
                    (buf + (((half * 2 + dz) * 16) + (yo * 4 + dy)) * 16 + xo * 4);
                m.x = fmaxf(m.x, v.x);
                m.y = fmaxf(m.y, v.y);
                m.z = fmaxf(m.z, v.z);
                m.w = fmaxf(m.w, v.w);
            }
        }
        float v = fmaxf(fmaxf(m.x, m.y), fmaxf(m.z, m.w));
        // combine the two z-halves across the wave (lane L <-> L^16)
        v = fmaxf(v, __shfl_xor(v, 16, 32));

        if (lane < 16) {
            const int c  = c0 + (t >> 2);
            const int zo = t & 3;
            // out[n][c][zo][yo][xo] ; 16 consecutive floats per half-wave
            out[(((size_t)n * CCH + c) * 4 + zo) * 16 + oi] = v;
        }
    }
}

extern "C" void kernel_launch(void* const* d_in, const int* in_sizes, int n_in,
                              void* d_out, int out_size, void* d_ws, size_t ws_size,
                              hipStream_t stream) {
    const float* fm    = (const float*)d_in[0];
    const int*   props = (const int*)d_in[1];
    float*       out   = (float*)d_out;
    const int N = in_sizes[1] / 7;   // proposals are [N,7] int32
    roi_maxpool3d_async_kernel<<<N, 256, 0, stream>>>(fm, props, out);
}